// DistanceKMeanLoss_46557445488919
// MI455X (gfx1250) — compile-verified
//
#include <hip/hip_runtime.h>
#include <math.h>

// knn_mean_distance on MI455X (gfx1250, wave32).
// d2 tiles computed with V_WMMA_F32_16X16X4_F32 via K=4 coordinate augmentation:
//   A(16x4)  = candidate [x, y, z, ||c||^2]      (M axis = candidates)
//   B(4x16)  = query     [-2x, -2y, -2z, 1]      (N axis = queries)
//   C(16x16) = ||q||^2 broadcast per column
// => D[r][lane] = ||c||^2 - 2 q.c + ||q||^2, with each lane holding 8
// candidate distances of ONE query row (selection stays register-local).
// Sorted-17 insertion is a med3 chain: h[t] = med3(v, h[t-1], h[t]).

typedef __attribute__((ext_vector_type(2))) float v2f;
typedef __attribute__((ext_vector_type(4))) float v4f;
typedef __attribute__((ext_vector_type(8))) float v8f;

#define NPTS 4096
#define NBATCH 16
#define KSEL 17      // k+1 smallest kept, smallest (self) dropped
#define BLK 256      // 8 waves; wave handles 16 query rows -> 128 rows/block
#define FMAXV 3.402823466e38f

__global__ __launch_bounds__(BLK) void knn_topk_kernel(
    const float* __restrict__ pcs, float* __restrict__ partial) {
  __shared__ __align__(16) float smem[NPTS * 4];   // 64 KB: float4 per point
  v4f* pts = (v4f*)smem;

  const int tid  = threadIdx.x;
  const int lane = tid & 31;
  const int wave = tid >> 5;
  const int half = lane >> 4;   // 0: K=0,1 lanes   1: K=2,3 lanes
  const int qcol = lane & 15;   // query column / candidate row within tile

  const int b  = blockIdx.x >> 5;          // 32 blocks per batch
  const int rb = (blockIdx.x & 31) << 7;   // 128 query rows per block

  // ---- cooperative load + repack: [x,y,z] -> float4(x,y,z,||p||^2) ----
  const float* pb = pcs + (size_t)b * NPTS * 3;
  for (int j = tid; j < NPTS; j += BLK) {
    float x = pb[3 * j + 0], y = pb[3 * j + 1], z = pb[3 * j + 2];
    v4f p; p.x = x; p.y = y; p.z = z; p.w = fmaf(x, x, fmaf(y, y, z * z));
    pts[j] = p;
  }
  __syncthreads();

  // ---- per-wave query setup (constant for all 256 column tiles) ----
  const int q = rb + wave * 16 + qcol;
  v4f qp = pts[q];
  v2f bfrag;
  if (half == 0) { bfrag.x = -2.0f * qp.x; bfrag.y = -2.0f * qp.y; }
  else           { bfrag.x = -2.0f * qp.z; bfrag.y = 1.0f;        }
  const float sqq = qp.w;
  v8f cacc;
  #pragma unroll
  for (int r = 0; r < 8; ++r) cacc[r] = sqq;   // per-column ||q||^2 preload

  // ---- sorted 17-smallest per lane, kept entirely in registers ----
  float h[KSEL];
  #pragma unroll
  for (int t = 0; t < KSEL; ++t) h[t] = FMAXV;

  // A-fragment address for this lane: candidate jb+qcol, half selects
  // (x,y) for lanes 0-15 or (z,||c||^2) for lanes 16-31.
  const float* abase = (const float*)pts + qcol * 4 + half * 2;

  // Software pipeline: prefetch next tile's A fragment during selection.
  // Index wraps (power of two) so the steady-state loop has no tail branch;
  // the final prefetch harmlessly re-reads tile 0.
  v2f a_cur = *(const v2f*)(abase);                    // ds_load_b64 (jb=0)
  for (int jb = 0; jb < NPTS; jb += 16) {
    v2f a_nxt = *(const v2f*)(abase + (((jb + 16) & (NPTS - 1)) * 4));

    v8f d = __builtin_amdgcn_wmma_f32_16x16x4_f32(
        false, a_cur, false, bfrag, (short)0, cacc, false, false);

    #pragma unroll
    for (int r = 0; r < 8; ++r) {
      // clamp to [0, FLT_MAX] in one v_med3 (no canonicalizing maxnum)
      float v = __builtin_amdgcn_fmed3f(d[r], 0.0f, FMAXV);
      // wave-wide skip: only run the insert chain if some lane qualifies
      if (__builtin_amdgcn_ballot_w32(v < h[16])) {
        // sorted insert; med3 is a no-op for lanes with v >= h[16]
        #pragma unroll
        for (int t = 16; t >= 1; --t)
          h[t] = __builtin_amdgcn_fmed3f(v, h[t - 1], h[t]);
        h[0] = __builtin_amdgcn_fmed3f(v, -FMAXV, h[0]);   // min(v, h[0])
      }
    }
    a_cur = a_nxt;
  }

  // ---- merge the two half-lists per query row (reuse LDS after barrier) ----
  __syncthreads();
  #pragma unroll
  for (int t = 0; t < KSEL; ++t) smem[tid * KSEL + t] = h[t];
  __syncthreads();

  float s = 0.0f;
  if (lane < 16) {
    const float* La = &smem[(wave * 32 + lane) * KSEL];
    const float* Lb = &smem[(wave * 32 + lane + 16) * KSEL];
    int ia = 0, ib = 0;
    for (int t = 0; t < KSEL; ++t) {        // ia+ib == t <= 16: in bounds
      float va = La[ia], vb = Lb[ib];
      float m;
      if (va <= vb) { m = va; ++ia; } else { m = vb; ++ib; }
      if (t > 0) s += sqrtf(fmaxf(m, 1e-12f));   // t==0 drops self-distance
    }
  }
  __syncthreads();
  if (lane < 16) smem[wave * 16 + lane] = s;
  __syncthreads();
  if (tid == 0) {
    float tot = 0.0f;
    for (int i = 0; i < 128; ++i) tot += smem[i];
    partial[blockIdx.x] = tot;
  }
}

// Deterministic fixed-order final reduction (no float atomics).
__global__ void knn_finalize_kernel(const float* __restrict__ partial,
                                    float* __restrict__ out, int n, float scale) {
  if (blockIdx.x == 0 && threadIdx.x == 0) {
    float t = 0.0f;
    for (int i = 0; i < n; ++i) t += partial[i];
    out[0] = t * scale;
  }
}

extern "C" void kernel_launch(void* const* d_in, const int* in_sizes, int n_in,
                              void* d_out, int out_size, void* d_ws, size_t ws_size,
                              hipStream_t stream) {
  (void)in_sizes; (void)n_in; (void)out_size; (void)ws_size;
  const float* pcs = (const float*)d_in[0];      // [16, 4096, 3] f32
  // d_in[1] is k == 16 (compiled in).
  float* part = (float*)d_ws;                    // 512 block partials
  float* out  = (float*)d_out;

  const int nblocks = NBATCH * (NPTS / 128);     // 512
  knn_topk_kernel<<<nblocks, BLK, 0, stream>>>(pcs, part);
  // mean over B*N*k = 16*4096*16 = 1048576 sqrt terms
  knn_finalize_kernel<<<1, 32, 0, stream>>>(part, out, nblocks,
                                            1.0f / 1048576.0f);
}